// QuantumAttention_86706799772085
// MI455X (gfx1250) — compile-verified
//
#include <hip/hip_runtime.h>
#include <math.h>

// ---------------------------------------------------------------------------
// QuantumAttention on MI455X (gfx1250, wave32).
//
// Analytic factorization: the 8-qubit score circuit is psi_A(x) (wires 0-3)
// tensor phi_k (wires 4-7, sample-independent) until the CNOT(i,4+i) layer,
// after which  expZ(4+j) = Zq_j(sample) * Zk_j(const),  expX(4+j) = Xk_j(const).
// Everything reduces to two 4-qubit sims per sample whose dominant cost is a
// fixed 16x16 complex unitary applied to 16 per-sample state vectors
// => V_WMMA_F32_16X16X4_F32 (full fp32).
//
// The 96->4 input projection is ALSO done with WMMA: pre_w is padded to a
// 16x96 A-matrix with W in rows 0-3 and duplicated into rows 8-11, so both
// lane-halves of the wave receive xq/xk in D[0..3] without any shuffles.
//
// Cross-half (xor-16) exchanges use v_permlanex16_b32 (pure VALU, no LDS).
// Setup kernel (runs once, 33 threads) is compiled with unrolling disabled to
// keep its static footprint small.
// ---------------------------------------------------------------------------

typedef __attribute__((ext_vector_type(2))) float v2f;
typedef __attribute__((ext_vector_type(8))) float v8f;

#define OFF_UQR 0
#define OFF_UQI 256
#define OFF_UQM 512   // -imag(Uq)  (f32 WMMA has no A-negate -> precompute)
#define OFF_UVR 768
#define OFF_UVI 1024
#define OFF_UVM 1280
#define OFF_ZK  1536
#define OFF_XK  1540
#define OFF_WP  1552  // padded 16x96 projection weights
#define WS_FLOATS (OFF_WP + 16 * 96)

// ============================ setup kernel =================================
// Builds the fixed 16x16 unitaries U_q, U_v (one basis column per thread),
// the constants Zk[4], Xk[4] from phi_k = U_k|0000>, and the padded W.
// Runs once; loops intentionally NOT unrolled (small static code size).

__device__ void g_rz16(float* vr, float* vi, int w, float t) {
  float c = cosf(0.5f * t), s = sinf(0.5f * t);
  int bit = 8 >> w;
#pragma clang loop unroll(disable)
  for (int m = 0; m < 16; m++) {
    float r = vr[m], im = vi[m];
    if (m & bit) { vr[m] = c * r - s * im; vi[m] = c * im + s * r; }   // e^{+it/2}
    else         { vr[m] = c * r + s * im; vi[m] = c * im - s * r; }   // e^{-it/2}
  }
}

__device__ void g_rx16(float* vr, float* vi, int w, float t) {
  float c = cosf(0.5f * t), s = sinf(0.5f * t);
  int bit = 8 >> w;
#pragma clang loop unroll(disable)
  for (int m = 0; m < 16; m++) {
    if (m & bit) continue;
    int b = m | bit;
    float ar = vr[m], ai = vi[m], br = vr[b], bi = vi[b];
    vr[m] = c * ar + s * bi;  vi[m] = c * ai - s * br;
    vr[b] = c * br + s * ai;  vi[b] = c * bi - s * ar;
  }
}

__device__ void g_crx16(float* vr, float* vi, int cw, int tw, float t) {
  float c = cosf(0.5f * t), s = sinf(0.5f * t);
  int bc = 8 >> cw, bt = 8 >> tw;
#pragma clang loop unroll(disable)
  for (int m = 0; m < 16; m++) {
    if (!(m & bc) || (m & bt)) continue;
    int b = m | bt;
    float ar = vr[m], ai = vi[m], br = vr[b], bi = vi[b];
    vr[m] = c * ar + s * bi;  vi[m] = c * ai - s * br;
    vr[b] = c * br + s * ai;  vi[b] = c * bi - s * ar;
  }
}

__device__ void g_cnot16(float* vr, float* vi, int cw, int tw) {
  int bc = 8 >> cw, bt = 8 >> tw;
#pragma clang loop unroll(disable)
  for (int m = 0; m < 16; m++) {
    if (!(m & bc) || (m & bt)) continue;
    int b = m | bt;
    float tr = vr[m]; vr[m] = vr[b]; vr[b] = tr;
    float ti = vi[m]; vi[m] = vi[b]; vi[b] = ti;
  }
}

__device__ void layer16(float* vr, float* vi, const float* rot, const float* crx) {
#pragma clang loop unroll(disable)
  for (int j = 0; j < 4; j++) {
    g_rz16(vr, vi, j, rot[3 * j]);
    g_rx16(vr, vi, j, rot[3 * j + 1]);
    g_rz16(vr, vi, j, rot[3 * j + 2]);
  }
  const int ra[8] = {0, 1, 2, 3, 0, 1, 2, 3};
  const int rb[8] = {1, 2, 3, 0, 3, 0, 1, 2};
#pragma clang loop unroll(disable)
  for (int k = 0; k < 8; k++) g_crx16(vr, vi, ra[k], rb[k], crx[k]);
#pragma clang loop unroll(disable)
  for (int k = 0; k < 8; k++) g_cnot16(vr, vi, ra[k], rb[k]);
#pragma clang loop unroll(disable)
  for (int j = 0; j < 4; j++) {
    g_rz16(vr, vi, j, rot[12 + 3 * j]);
    g_rx16(vr, vi, j, rot[13 + 3 * j]);
    g_rz16(vr, vi, j, rot[14 + 3 * j]);
  }
}

__global__ void qa_setup(const float* __restrict__ q_rot, const float* __restrict__ q_crx,
                         const float* __restrict__ k_rot, const float* __restrict__ k_crx,
                         const float* __restrict__ v_rot, const float* __restrict__ v_crx,
                         const float* __restrict__ pre_w, float* __restrict__ ws) {
  int t = threadIdx.x;
  // Padded projection weights: rows 0-3 = W, rows 8-11 = W (so both wave
  // halves receive the projection in D[0..3]), everything else 0.
#pragma clang loop unroll(disable)
  for (int i = t; i < 16 * 96; i += 256) {
    int row = i / 96, col = i - row * 96;
    float v = 0.f;
    if (row < 4)                   v = pre_w[row * 96 + col];
    else if (row >= 8 && row < 12) v = pre_w[(row - 8) * 96 + col];
    ws[OFF_WP + i] = v;
  }
  if (t >= 33) return;
  float vr[16], vi[16];
  const float* rot;
  const float* crx;
  int col;
  if (t < 16)      { rot = q_rot; crx = q_crx; col = t; }
  else if (t < 32) { rot = v_rot; crx = v_crx; col = t - 16; }
  else             { rot = k_rot; crx = k_crx; col = 0; }
#pragma clang loop unroll(disable)
  for (int m = 0; m < 16; m++) { vr[m] = (m == col) ? 1.f : 0.f; vi[m] = 0.f; }
  layer16(vr, vi, rot, crx);
  if (t < 16) {
#pragma clang loop unroll(disable)
    for (int m = 0; m < 16; m++) {
      ws[OFF_UQR + m * 16 + col] = vr[m];
      ws[OFF_UQI + m * 16 + col] = vi[m];
      ws[OFF_UQM + m * 16 + col] = -vi[m];
    }
  } else if (t < 32) {
#pragma clang loop unroll(disable)
    for (int m = 0; m < 16; m++) {
      ws[OFF_UVR + m * 16 + col] = vr[m];
      ws[OFF_UVI + m * 16 + col] = vi[m];
      ws[OFF_UVM + m * 16 + col] = -vi[m];
    }
  } else {
#pragma clang loop unroll(disable)
    for (int j = 0; j < 4; j++) {
      int bit = 8 >> j;
      float z = 0.f, x = 0.f;
#pragma clang loop unroll(disable)
      for (int m = 0; m < 16; m++) {
        float p = vr[m] * vr[m] + vi[m] * vi[m];
        z += (m & bit) ? -p : p;
        if (!(m & bit)) {
          int b = m | bit;
          x += vr[m] * vr[b] + vi[m] * vi[b];
        }
      }
      ws[OFF_ZK + j] = z;
      ws[OFF_XK + j] = 2.f * x;
    }
  }
}

// ============================= main kernel =================================

// xor-16 cross-half exchange via v_permlanex16_b32 (identity lane selects).
__device__ __forceinline__ float sxh(float v) {
  unsigned u = __float_as_uint(v);
  unsigned r = __builtin_amdgcn_permlanex16(u, u, 0x76543210u, 0xFEDCBA98u, false, false);
  return __uint_as_float(r);
}

// RX on wire 0 (amplitude bit 8 == lane-half bit): cross-half exchange.
__device__ __forceinline__ void rx_wire0(float* sr, float* si, float c, float s) {
#pragma unroll
  for (int r = 0; r < 8; r++) {
    float pr = sxh(sr[r]), pi = sxh(si[r]);
    float nr = c * sr[r] + s * pi;
    float ni = c * si[r] - s * pr;
    sr[r] = nr; si[r] = ni;
  }
}

// RX on wires 1..3 (stride d = 4,2,1 inside the 8 per-lane amplitudes).
__device__ __forceinline__ void rx_inner(float* sr, float* si, int d, float c, float s) {
#pragma unroll
  for (int a = 0; a < 8; a++) {
    if (a & d) continue;
    int b = a | d;
    float ar = sr[a], ai = si[a], br = sr[b], bi = si[b];
    sr[a] = c * ar + s * bi;  si[a] = c * ai - s * br;
    sr[b] = c * br + s * ai;  si[b] = c * bi - s * ar;
  }
}

__device__ __forceinline__ void embed_rx(float* sr, float* si, const float* cc, const float* ss) {
  rx_wire0(sr, si, cc[0], ss[0]);
  rx_inner(sr, si, 4, cc[1], ss[1]);
  rx_inner(sr, si, 2, cc[2], ss[2]);
  rx_inner(sr, si, 1, cc[3], ss[3]);
}

// Product-state init: amp[m] = prod_i (bit_i(m) ? -i*s_i : c_i) applied to
// |0000>. Phase is (-i)^popcount; q = popcount of the r-bits (compile time),
// plus h (runtime select).
__device__ __forceinline__ void init_embed(float* sr, float* si,
                                           const float* c, const float* s, int h) {
  float f0 = h ? s[0] : c[0];
#pragma unroll
  for (int r = 0; r < 8; r++) {
    float f1 = (r & 4) ? s[1] : c[1];
    float f2 = (r & 2) ? s[2] : c[2];
    float f3 = (r & 1) ? s[3] : c[3];
    float mag = f0 * f1 * f2 * f3;
    const int q = ((r >> 2) & 1) + ((r >> 1) & 1) + (r & 1);
    if (q == 0)      { sr[r] = h ? 0.f : mag;   si[r] = h ? -mag : 0.f; }
    else if (q == 1) { sr[r] = h ? -mag : 0.f;  si[r] = h ? 0.f : -mag; }
    else if (q == 2) { sr[r] = h ? 0.f : -mag;  si[r] = h ? mag : 0.f; }
    else             { sr[r] = h ? mag : 0.f;   si[r] = h ? 0.f : mag; }
  }
}

// A-matrix chunk c (K = 4c..4c+3) of a 16x16 row-major matrix:
// VGPR0 = U[m][4c+2h], VGPR1 = U[m][4c+2h+1].
__device__ __forceinline__ v2f loadA16(const float* __restrict__ U, int m, int h, int c) {
  return *(const v2f*)(U + m * 16 + 4 * c + 2 * h);
}

// Build the 4 B-operand chunks (4x16, K x samples) from the state registers.
__device__ __forceinline__ void buildB(const float* s, int h, v2f* B) {
  float t0, t1;
  t0 = sxh(s[2]); t1 = sxh(s[3]);
  B[0][0] = h ? t0 : s[0];  B[0][1] = h ? t1 : s[1];     // K=0,2 / 1,3
  t0 = sxh(s[6]); t1 = sxh(s[7]);
  B[1][0] = h ? t0 : s[4];  B[1][1] = h ? t1 : s[5];     // K=4,6 / 5,7
  t0 = sxh(s[0]); t1 = sxh(s[1]);
  B[2][0] = h ? s[2] : t0;  B[2][1] = h ? s[3] : t1;     // K=8,10 / 9,11
  t0 = sxh(s[4]); t1 = sxh(s[5]);
  B[3][0] = h ? s[6] : t0;  B[3][1] = h ? s[7] : t1;     // K=12,14 / 13,15
}

// psi <- U * psi  (complex), via 16 x V_WMMA_F32_16X16X4_F32.
// Dr = Ur*Br + (-Ui)*Bi ; Di = Ur*Bi + Ui*Br. D layout == state layout.
__device__ __forceinline__ void applyU(float* sr, float* si,
                                       const float* __restrict__ Ur,
                                       const float* __restrict__ Ui,
                                       const float* __restrict__ Um,
                                       int m, int h) {
  v2f Br[4], Bi[4];
  buildB(sr, h, Br);
  buildB(si, h, Bi);
  v8f Dr = {0.f, 0.f, 0.f, 0.f, 0.f, 0.f, 0.f, 0.f};
  v8f Di = {0.f, 0.f, 0.f, 0.f, 0.f, 0.f, 0.f, 0.f};
#pragma unroll
  for (int c = 0; c < 4; c++) {
    v2f ar = loadA16(Ur, m, h, c);
    v2f ai = loadA16(Ui, m, h, c);
    v2f am = loadA16(Um, m, h, c);
    Dr = __builtin_amdgcn_wmma_f32_16x16x4_f32(false, ar, false, Br[c], (short)0, Dr, false, false);
    Dr = __builtin_amdgcn_wmma_f32_16x16x4_f32(false, am, false, Bi[c], (short)0, Dr, false, false);
    Di = __builtin_amdgcn_wmma_f32_16x16x4_f32(false, ar, false, Bi[c], (short)0, Di, false, false);
    Di = __builtin_amdgcn_wmma_f32_16x16x4_f32(false, ai, false, Br[c], (short)0, Di, false, false);
  }
#pragma unroll
  for (int r = 0; r < 8; r++) { sr[r] = Dr[r]; si[r] = Di[r]; }
}

__device__ __forceinline__ void measureZ(const float* sr, const float* si, int h, float* z) {
  float p[8];
#pragma unroll
  for (int r = 0; r < 8; r++) p[r] = sr[r] * sr[r] + si[r] * si[r];
  float z0 = p[0] + p[1] + p[2] + p[3] + p[4] + p[5] + p[6] + p[7];
  z0 = h ? -z0 : z0;
  float z1 = p[0] + p[1] + p[2] + p[3] - p[4] - p[5] - p[6] - p[7];
  float z2 = p[0] + p[1] - p[2] - p[3] + p[4] + p[5] - p[6] - p[7];
  float z3 = p[0] - p[1] + p[2] - p[3] + p[4] - p[5] + p[6] - p[7];
  z[0] = z0 + sxh(z0);
  z[1] = z1 + sxh(z1);
  z[2] = z2 + sxh(z2);
  z[3] = z3 + sxh(z3);
}

__device__ __forceinline__ void measureX(const float* sr, const float* si, float* x) {
  float t0 = 0.f;
#pragma unroll
  for (int r = 0; r < 8; r++) t0 += sr[r] * sxh(sr[r]) + si[r] * sxh(si[r]);
  x[0] = 2.f * t0;   // both halves compute identical full sum
  float t1 = sr[0] * sr[4] + si[0] * si[4] + sr[1] * sr[5] + si[1] * si[5]
           + sr[2] * sr[6] + si[2] * si[6] + sr[3] * sr[7] + si[3] * si[7];
  float t2 = sr[0] * sr[2] + si[0] * si[2] + sr[1] * sr[3] + si[1] * si[3]
           + sr[4] * sr[6] + si[4] * si[6] + sr[5] * sr[7] + si[5] * si[7];
  float t3 = sr[0] * sr[1] + si[0] * si[1] + sr[2] * sr[3] + si[2] * si[3]
           + sr[4] * sr[5] + si[4] * si[5] + sr[6] * sr[7] + si[6] * si[7];
  x[1] = 2.f * (t1 + sxh(t1));
  x[2] = 2.f * (t2 + sxh(t2));
  x[3] = 2.f * (t3 + sxh(t3));
}

// CNOT ring (0,1),(1,2),(2,3),(3,0) on the 4-qubit state.
__device__ __forceinline__ void cnot_ring(float* sr, float* si, int h) {
  // (0,1): control bit8 (= h), target bit4 -> h==1 lanes swap r <-> r^4
#pragma unroll
  for (int j = 0; j < 4; j++) {
    float a = sr[j], b = sr[j + 4];
    sr[j] = h ? b : a;  sr[j + 4] = h ? a : b;
    float c = si[j], d = si[j + 4];
    si[j] = h ? d : c;  si[j + 4] = h ? c : d;
  }
  // (1,2): control bit4, target bit2 -> swap (4,6),(5,7)
  { float t = sr[4]; sr[4] = sr[6]; sr[6] = t; t = si[4]; si[4] = si[6]; si[6] = t; }
  { float t = sr[5]; sr[5] = sr[7]; sr[7] = t; t = si[5]; si[5] = si[7]; si[7] = t; }
  // (2,3): control bit2, target bit1 -> swap (2,3),(6,7)
  { float t = sr[2]; sr[2] = sr[3]; sr[3] = t; t = si[2]; si[2] = si[3]; si[3] = t; }
  { float t = sr[6]; sr[6] = sr[7]; sr[7] = t; t = si[6]; si[6] = si[7]; si[7] = t; }
  // (3,0): control bit1, target bit8 -> odd r: cross-half swap
#pragma unroll
  for (int r = 1; r < 8; r += 2) { sr[r] = sxh(sr[r]); si[r] = sxh(si[r]); }
}

__global__ __launch_bounds__(256) void qa_main(
    const float* __restrict__ x1, const float* __restrict__ x2,
    const float* __restrict__ pre_b,
    const float* __restrict__ ln_g, const float* __restrict__ ln_b,
    const float* __restrict__ ws, float* __restrict__ out) {
  const int lane = threadIdx.x & 31;
  const int wave = threadIdx.x >> 5;
  const int n = lane & 15;        // sample within wave group / WMMA column
  const int h = lane >> 4;        // half: amplitude bit 8
  const int g = blockIdx.x * 8 + wave;
  const int s = g * 16 + n;       // global sample
  const int m = n;                // WMMA A-matrix row (both halves carry M=0-15)

  // ---- pre-projection via WMMA: D = Wpad(16x96) * X(96x16) ----
  // Wpad rows 0-3 = W and rows 8-11 = W, so D[0..3] carries the projection in
  // BOTH lane halves (rows 8-11 are what lanes 16-31 hold). No shuffles.
  const float* Wp = ws + OFF_WP;
  const float* rowq = x1 + (size_t)s * 96;
  const float* rowk = x2 + (size_t)s * 96;
  v8f Dq = {0.f, 0.f, 0.f, 0.f, 0.f, 0.f, 0.f, 0.f};
  v8f Dk = {0.f, 0.f, 0.f, 0.f, 0.f, 0.f, 0.f, 0.f};
#pragma unroll
  for (int c = 0; c < 24; c++) {
    const int off = 4 * c + 2 * h;
    v2f a  = *(const v2f*)(Wp + m * 96 + off);
    v2f bq = *(const v2f*)(rowq + off);
    v2f bk = *(const v2f*)(rowk + off);
    Dq = __builtin_amdgcn_wmma_f32_16x16x4_f32(false, a, false, bq, (short)0, Dq, false, false);
    Dk = __builtin_amdgcn_wmma_f32_16x16x4_f32(false, a, false, bk, (short)0, Dk, false, false);
  }
  float cq[4], sq[4], ck[4], sk[4];
#pragma unroll
  for (int i = 0; i < 4; i++) {
    float xq = Dq[i] + pre_b[i];
    float xk = Dk[i] + pre_b[i];
    cq[i] = __cosf(0.5f * xq); sq[i] = __sinf(0.5f * xq);
    ck[i] = __cosf(0.5f * xk); sk[i] = __sinf(0.5f * xk);
  }

  // ---- score circuit: psi_A = RX(xk) . U_q . RX(xq) |0000>, then Zq ----
  float sr[8], si[8];
  init_embed(sr, si, cq, sq, h);
  applyU(sr, si, ws + OFF_UQR, ws + OFF_UQI, ws + OFF_UQM, m, h);
  embed_rx(sr, si, ck, sk);
  float zq[4];
  measureZ(sr, si, h, zq);

  // ---- mag_j = sqrt((Zq_j*Zk_j)^2 + Xk_j^2), angle = tanh(mag)*pi ----
  float cm[4], sm[4];
#pragma unroll
  for (int j = 0; j < 4; j++) {
    float zk = ws[OFF_ZK + j];
    float xkc = ws[OFF_XK + j];
    float sz = zq[j] * zk;
    float mag = sqrtf(sz * sz + xkc * xkc);
    float ang = tanhf(mag) * 3.14159265358979323846f;
    cm[j] = __cosf(0.5f * ang);
    sm[j] = __sinf(0.5f * ang);
  }

  // ---- value circuit ----
  init_embed(sr, si, ck, sk, h);
  applyU(sr, si, ws + OFF_UVR, ws + OFF_UVI, ws + OFF_UVM, m, h);
  embed_rx(sr, si, cm, sm);
  cnot_ring(sr, si, h);
  float vz[4], vx[4];
  measureZ(sr, si, h, vz);
  measureX(sr, si, vx);

  // ---- LayerNorm(8) + ReLU ----
  float mu = (vz[0] + vz[1] + vz[2] + vz[3] + vx[0] + vx[1] + vx[2] + vx[3]) * 0.125f;
  float var = 0.f;
#pragma unroll
  for (int j = 0; j < 4; j++) {
    float a = vz[j] - mu, b = vx[j] - mu;
    var += a * a + b * b;
  }
  var *= 0.125f;
  float inv = rsqrtf(var + 1e-5f);
  float4 o;
#pragma unroll
  for (int i = 0; i < 4; i++) {
    float v = h ? vx[i] : vz[i];
    float y = (v - mu) * inv * ln_g[4 * h + i] + ln_b[4 * h + i];
    ((float*)&o)[i] = fmaxf(y, 0.f);
  }
  *(float4*)(out + (size_t)s * 8 + (size_t)h * 4) = o;
}

// ============================ launch =======================================
extern "C" void kernel_launch(void* const* d_in, const int* in_sizes, int n_in,
                              void* d_out, int out_size, void* d_ws, size_t ws_size,
                              hipStream_t stream) {
  const float* x1    = (const float*)d_in[0];
  const float* x2    = (const float*)d_in[1];
  const float* pre_w = (const float*)d_in[2];
  const float* pre_b = (const float*)d_in[3];
  const float* q_rot = (const float*)d_in[4];
  const float* q_crx = (const float*)d_in[5];
  const float* k_rot = (const float*)d_in[6];
  const float* k_crx = (const float*)d_in[7];
  const float* v_rot = (const float*)d_in[8];
  const float* v_crx = (const float*)d_in[9];
  const float* ln_g  = (const float*)d_in[10];
  const float* ln_b  = (const float*)d_in[11];
  float* ws  = (float*)d_ws;
  float* out = (float*)d_out;

  qa_setup<<<1, 256, 0, stream>>>(q_rot, q_crx, k_rot, k_crx, v_rot, v_crx, pre_w, ws);
  // 32768 samples / (16 per wave * 8 waves per block) = 256 blocks
  qa_main<<<256, 256, 0, stream>>>(x1, x2, pre_b, ln_g, ln_b, ws, out);
}